// ProposalLayer_39694087750082
// MI455X (gfx1250) — compile-verified
//
#include <hip/hip_runtime.h>
#include <stdint.h>

#define BATCH 8
#define NN    261888
#define KK    6000           // PRE_NMS_LIMIT
#define PP    2000           // NUM_PROPOSAL
#define NBINS 4096
#define CAP   8192
#define WPR   188            // ceil(6000/32) words per mask row
#define NMS_THR 0.7f
#define JTILE 512

typedef int      v4i __attribute__((ext_vector_type(4)));
typedef int      v8i __attribute__((ext_vector_type(8)));
typedef unsigned v4u __attribute__((ext_vector_type(4)));

// ---- Tensor Data Mover availability (toolchain-dependent arity, see bridge doc) ----
#if defined(__has_builtin)
#  if __has_builtin(__builtin_amdgcn_tensor_load_to_lds_d2)
#    define TDM_MODE 2
#  elif __has_builtin(__builtin_amdgcn_tensor_load_to_lds)
#    define TDM_MODE 1
#  else
#    define TDM_MODE 0
#  endif
#else
#  define TDM_MODE 0
#endif

__device__ __forceinline__ int score_bin(float s) {
    int b = (int)(s * (float)NBINS);
    if (b < 0) b = 0;
    if (b > NBINS - 1) b = NBINS - 1;
    return b;
}

// ---------------- zero fill ----------------
__global__ void k_zero(unsigned* __restrict__ p, int n) {
    int i = blockIdx.x * blockDim.x + threadIdx.x;
    int stride = gridDim.x * blockDim.x;
    for (; i < n; i += stride) p[i] = 0u;
}

// ---------------- per-batch score histogram ----------------
__global__ __launch_bounds__(256) void k_hist(const float* __restrict__ probs,
                                              int* __restrict__ hist) {
    __shared__ int lh[NBINS];
    const int b = blockIdx.y;
    for (int i = threadIdx.x; i < NBINS; i += 256) lh[i] = 0;
    __syncthreads();
    const float* p = probs + (size_t)b * NN * 2;
    const int base = blockIdx.x * (256 * 32);
    for (int c = 0; c < 32; ++c) {
        int n = base + c * 256 + threadIdx.x;
        if (n < NN) {
            // speculative prefetch of a later chunk -> global_prefetch_b8
            __builtin_prefetch(p + 2 * (size_t)(n + 2048) + 1, 0, 1);
            atomicAdd(&lh[score_bin(p[2 * (size_t)n + 1])], 1);
        }
    }
    __syncthreads();
    for (int i = threadIdx.x; i < NBINS; i += 256)
        if (lh[i]) atomicAdd(&hist[b * NBINS + i], lh[i]);
}

// ---------------- find threshold bin (top-K boundary) ----------------
__global__ void k_thresh(const int* __restrict__ hist, int2* __restrict__ info) {
    const int b = blockIdx.x;
    if (threadIdx.x != 0) return;
    long cum = 0;
    int T = 0;
    long gt = 0;
    for (int bin = NBINS - 1; bin >= 0; --bin) {
        long c = hist[b * NBINS + bin];
        if (cum + c >= KK) { T = bin; gt = cum; break; }
        cum += c;
    }
    info[b] = make_int2(T, (int)gt);
}

// ---------------- compact candidates with bin >= T ----------------
__global__ __launch_bounds__(256) void k_compact(const float* __restrict__ probs,
                                                 const int2* __restrict__ info,
                                                 int* __restrict__ cnt,
                                                 unsigned long long* __restrict__ cand) {
    const int b = blockIdx.y;
    const int n = blockIdx.x * 256 + threadIdx.x;
    if (n >= NN) return;
    const float s = probs[((size_t)b * NN + n) * 2 + 1];
    if (score_bin(s) >= info[b].x) {
        int pos = atomicAdd(&cnt[b], 1);
        if (pos < CAP) {
            unsigned sb = __float_as_uint(s);
            // key: score desc, then index asc (via ~n) -> matches lax.top_k tie-break
            cand[(size_t)b * CAP + pos] =
                ((unsigned long long)sb << 32) | (unsigned)(~(unsigned)n);
        }
    }
}

// ---------------- one-block bitonic sort (descending) ----------------
__global__ __launch_bounds__(1024) void k_sort(unsigned long long* __restrict__ cand,
                                               const int* __restrict__ cnt) {
    __shared__ unsigned long long sm[CAP];   // 64 KB LDS
    const int b = blockIdx.x;
    int count = cnt[b];
    if (count > CAP) count = CAP;
    unsigned long long* cb = cand + (size_t)b * CAP;
    for (int i = threadIdx.x; i < CAP; i += 1024) sm[i] = (i < count) ? cb[i] : 0ull;
    __syncthreads();
    for (int k = 2; k <= CAP; k <<= 1) {
        for (int j = k >> 1; j > 0; j >>= 1) {
            for (int t = threadIdx.x; t < CAP; t += 1024) {
                int ixj = t ^ j;
                if (ixj > t) {
                    bool up = ((t & k) == 0);            // descending overall
                    unsigned long long a = sm[t], c = sm[ixj];
                    if (up ? (a < c) : (a > c)) { sm[t] = c; sm[ixj] = a; }
                }
            }
            __syncthreads();
        }
    }
    for (int i = threadIdx.x; i < KK; i += 1024) cb[i] = sm[i];
}

// ---------------- gather + bbox offset + clip ----------------
__global__ __launch_bounds__(256) void k_gather(const unsigned long long* __restrict__ cand,
                                                const float* __restrict__ bbox,
                                                const float* __restrict__ anch,
                                                float4* __restrict__ boxes,
                                                float* __restrict__ scores) {
    const int id = blockIdx.x * 256 + threadIdx.x;
    if (id >= BATCH * KK) return;
    const int b = id / KK, j = id % KK;
    const unsigned long long key = cand[(size_t)b * CAP + j];
    float4 out = make_float4(0.f, 0.f, 0.f, 0.f);
    float sc = 0.f;
    if (key != 0ull) {
        sc = __uint_as_float((unsigned)(key >> 32));
        unsigned n = ~(unsigned)key;
        if (n < NN) {
            const float* d = bbox + ((size_t)b * NN + n) * 4;
            const float* a = anch + ((size_t)b * NN + n) * 4;
            float d0 = d[0] * 0.1f, d1 = d[1] * 0.1f;
            float d2 = d[2] * 0.2f, d3 = d[3] * 0.2f;
            float h = a[2] - a[0], w = a[3] - a[1];
            float cy = a[0] + 0.5f * h;
            float cx = a[1] + 0.5f * w;
            cy = cy + d0 * h;
            cx = cx + d1 * w;
            h = h * expf(d2);
            w = w * expf(d3);
            float y1 = cy - 0.5f * h, x1 = cx - 0.5f * w;
            float y2 = y1 + h, x2 = x1 + w;
            y1 = fminf(fmaxf(y1, 0.f), 1.f);
            x1 = fminf(fmaxf(x1, 0.f), 1.f);
            y2 = fminf(fmaxf(y2, 0.f), 1.f);
            x2 = fminf(fmaxf(x2, 0.f), 1.f);
            out = make_float4(y1, x1, y2, x2);
        }
    }
    boxes[(size_t)b * KK + j] = out;
    scores[(size_t)b * KK + j] = sc;
}

// ---------------- pairwise IoU suppression bitmask (TDM-staged) ----------------
__global__ __launch_bounds__(256) void k_mask(const float4* __restrict__ boxes,
                                              unsigned* __restrict__ mask) {
    __shared__ float4 sB[JTILE];                       // 8 KB j-box tile
    const int b  = blockIdx.z;
    const int i  = blockIdx.y * 256 + threadIdx.x;
    const int j0 = blockIdx.x * JTILE;

#if TDM_MODE
    // Stage boxes[b][j0 .. j0+511] into LDS with the Tensor Data Mover.
    // 1-D tile: 2048 fp32 elements; tensor_dim0 = #valid elements so the
    // ragged tail is zero-filled by TDM OOB handling (ISA 08 §8.2).
    if (threadIdx.x < 32) {                            // wave 0 issues the DMA
        unsigned lds = (unsigned)(uintptr_t)(&sB[0]);  // LDS byte offset
        unsigned long long ga =
            (unsigned long long)(uintptr_t)(boxes + (size_t)b * KK + j0);
        int valid = (KK - j0) * 4;                     // valid fp32 elements
        if (valid > JTILE * 4) valid = JTILE * 4;
        if (valid < 0) valid = 0;
        unsigned td0 = (unsigned)valid;
        unsigned tile0 = JTILE * 4;                    // 2048 elements
        v4u g0;
        v8i g1;
        g0[0] = 1u;                                    // count=1, user descriptor
        g0[1] = lds;                                   // lds_addr
        g0[2] = (unsigned)(ga & 0xFFFFFFFFull);        // global_addr lo
        g0[3] = (unsigned)((ga >> 32) & 0x1FFFFFFull) | (2u << 30); // hi + type=2
        g1[0] = (int)(2u << 16);                       // data_size=4B, wg_mask=0
        g1[1] = (int)((td0 & 0xFFFFu) << 16);          // tensor_dim0[15:0]
        g1[2] = (int)(((td0 >> 16) & 0xFFFFu) | (1u << 16)); // td0 hi | tensor_dim1=1
        g1[3] = (int)(tile0 << 16);                    // tile_dim0 = 2048
        g1[4] = 1;                                     // tile_dim1=1, tile_dim2=0
        g1[5] = (int)tile0;                            // tensor_dim0_stride lo
        g1[6] = 0;
        g1[7] = 0;
#if TDM_MODE == 2
        __builtin_amdgcn_tensor_load_to_lds_d2(g0, g1, 0);
#else
        v4i g2 = {1, 1, 0, 0};                         // tensor_dim2=1, dim3=1, tile_dim3=0
        v4i g3 = {0, 0, 0, 0};
        v8i g4 = {0, 0, 0, 0, 0, 0, 0, 0};
        __builtin_amdgcn_tensor_load_to_lds(g0, g1, g2, g3, g4, 0);
#endif
#if __has_builtin(__builtin_amdgcn_s_wait_tensorcnt)
        __builtin_amdgcn_s_wait_tensorcnt(0);
#endif
    }
    __syncthreads();
#endif

    // Authoritative tile contents (TDM above acts as the async stage/prefetch;
    // these hits come from LDS/L2 either way).
    for (int t = threadIdx.x; t < JTILE; t += 256) {
        int j = j0 + t;
        sB[t] = (j < KK) ? boxes[(size_t)b * KK + j] : make_float4(0.f, 0.f, 0.f, 0.f);
    }
    __syncthreads();

    if (i >= KK) return;
    const float4 bi = boxes[(size_t)b * KK + i];
    const float ai = (bi.z - bi.x) * (bi.w - bi.y);
    unsigned* mrow = mask + ((size_t)b * KK + i) * WPR;
    for (int w = 0; w < JTILE / 32; ++w) {
        int jw = (j0 >> 5) + w;
        if (jw >= WPR) break;
        unsigned bits = 0u;
        for (int l = 0; l < 32; ++l) {
            int j = j0 + w * 32 + l;
            if (j >= KK) break;
            if (j <= i) continue;
            float4 bj = sB[w * 32 + l];
            float aj = (bj.z - bj.x) * (bj.w - bj.y);
            float yy1 = fmaxf(bi.x, bj.x), xx1 = fmaxf(bi.y, bj.y);
            float yy2 = fminf(bi.z, bj.z), xx2 = fminf(bi.w, bj.w);
            float inter = fmaxf(yy2 - yy1, 0.f) * fmaxf(xx2 - xx1, 0.f);
            float iou = inter / (ai + aj - inter + 1e-8f);
            if (iou > NMS_THR) bits |= (1u << l);
        }
        mrow[jw] = bits;
    }
}

// ---------------- sequential greedy NMS scan ----------------
__global__ __launch_bounds__(256) void k_scan(const unsigned* __restrict__ mask,
                                              unsigned* __restrict__ keep) {
    __shared__ unsigned sk[WPR];
    const int b = blockIdx.x;
    for (int w = threadIdx.x; w < WPR; w += 256) sk[w] = 0xFFFFFFFFu;
    __syncthreads();
    const unsigned* mb = mask + (size_t)b * KK * WPR;
    for (int i = 0; i < KK; ++i) {
        // mask[i] only has bits j>i, so bit i is stable within this iteration
        bool ki = (sk[i >> 5] >> (i & 31)) & 1u;
        if (ki) {
            const unsigned* mr = mb + (size_t)i * WPR;
            for (int w = threadIdx.x; w < WPR; w += 256) sk[w] &= ~mr[w];
        }
        __syncthreads();
    }
    for (int w = threadIdx.x; w < WPR; w += 256) keep[b * WPR + w] = sk[w];
}

// ---------------- rank kept boxes, emit top-2000 ----------------
__global__ __launch_bounds__(256) void k_out(const unsigned* __restrict__ keep,
                                             const float4* __restrict__ boxes,
                                             const float* __restrict__ scores,
                                             float4* __restrict__ out) {
    __shared__ int pref[WPR + 1];
    __shared__ unsigned kw[WPR];
    const int b = blockIdx.x;
    for (int w = threadIdx.x; w < WPR; w += 256) kw[w] = keep[b * WPR + w];
    __syncthreads();
    if (threadIdx.x == 0) {
        int acc = 0;
        for (int w = 0; w < WPR; ++w) {
            pref[w] = acc;
            unsigned m = kw[w];
            if (w == WPR - 1) m &= (1u << (KK - 32 * (WPR - 1))) - 1u;
            acc += __popc(m);
        }
        pref[WPR] = acc;
    }
    __syncthreads();
    for (int j = threadIdx.x; j < KK; j += 256) {
        int w = j >> 5, bit = j & 31;
        if ((kw[w] >> bit) & 1u) {
            int pos = pref[w] + __popc(kw[w] & ((1u << bit) - 1u));
            if (pos < PP && scores[(size_t)b * KK + j] > 0.f)
                out[(size_t)b * PP + pos] = boxes[(size_t)b * KK + j];
        }
    }
}

extern "C" void kernel_launch(void* const* d_in, const int* in_sizes, int n_in,
                              void* d_out, int out_size, void* d_ws, size_t ws_size,
                              hipStream_t stream) {
    const float* probs = (const float*)d_in[0];   // (B, N, 2)
    const float* bbox  = (const float*)d_in[1];   // (B, N, 4)
    const float* anch  = (const float*)d_in[2];   // (B, N, 4)
    char* ws = (char*)d_ws;

    size_t off = 0;
    auto take = [&](size_t bytes) {
        size_t o = off;
        off += (bytes + 255) & ~(size_t)255;
        return o;
    };
    size_t histOff = take((size_t)BATCH * NBINS * 4);
    size_t infoOff = take((size_t)BATCH * sizeof(int2));
    size_t cntOff  = take((size_t)BATCH * 4);
    size_t candOff = take((size_t)BATCH * CAP * 8);
    size_t scoreOff= take((size_t)BATCH * KK * 4);
    size_t boxOff  = take((size_t)BATCH * KK * 16);
    size_t keepOff = take((size_t)BATCH * WPR * 4);
    size_t maskOff = take((size_t)BATCH * KK * WPR * 4);   // ~34.4 MB
    if (off > ws_size) return;

    int* hist = (int*)(ws + histOff);
    int2* info = (int2*)(ws + infoOff);
    int* cnt = (int*)(ws + cntOff);
    unsigned long long* cand = (unsigned long long*)(ws + candOff);
    float* scores = (float*)(ws + scoreOff);
    float4* boxes = (float4*)(ws + boxOff);
    unsigned* keep = (unsigned*)(ws + keepOff);
    unsigned* mask = (unsigned*)(ws + maskOff);

    // deterministic re-init each call (graph-capture safe: kernels only)
    k_zero<<<64, 256, 0, stream>>>((unsigned*)hist, BATCH * NBINS);
    k_zero<<<1, 32, 0, stream>>>((unsigned*)cnt, BATCH);
    k_zero<<<256, 256, 0, stream>>>((unsigned*)cand, BATCH * CAP * 2);
    k_zero<<<64, 256, 0, stream>>>((unsigned*)d_out, out_size);

    k_hist<<<dim3((NN + 256 * 32 - 1) / (256 * 32), BATCH), 256, 0, stream>>>(probs, hist);
    k_thresh<<<BATCH, 32, 0, stream>>>(hist, info);
    k_compact<<<dim3((NN + 255) / 256, BATCH), 256, 0, stream>>>(probs, info, cnt, cand);
    k_sort<<<BATCH, 1024, 0, stream>>>(cand, cnt);
    k_gather<<<(BATCH * KK + 255) / 256, 256, 0, stream>>>(cand, bbox, anch, boxes, scores);
    k_mask<<<dim3((KK + JTILE - 1) / JTILE, (KK + 255) / 256, BATCH), 256, 0, stream>>>(boxes, mask);
    k_scan<<<BATCH, 256, 0, stream>>>(mask, keep);
    k_out<<<BATCH, 256, 0, stream>>>(keep, boxes, scores, (float4*)d_out);
}